// CausalSelfAttentionKV_56255481643728
// MI455X (gfx1250) — compile-verified
//
#include <hip/hip_runtime.h>
#include <math.h>

// Problem constants (from reference)
#define B_    16
#define T_    8
#define C_    2048
#define H_    16
#define D_    128
#define MAXS_ 4096
// M dimension of both GEMMs: B*T = 128 rows

typedef __attribute__((ext_vector_type(2))) float v2f;
typedef __attribute__((ext_vector_type(8))) float v8f;

__device__ __forceinline__ v8f wmma4(v2f a, v2f b, v8f c) {
  // D = A(16x4 f32) * B(4x16 f32) + C(16x16 f32)
  return __builtin_amdgcn_wmma_f32_16x16x4_f32(
      /*neg_a=*/false, a, /*neg_b=*/false, b,
      /*c_mod=*/(short)0, c, /*reuse_a=*/false, /*reuse_b=*/false);
}

__device__ __forceinline__ void lds_fence() {
  asm volatile("s_wait_dscnt 0" ::: "memory");
}

__device__ __forceinline__ void async_fence() {
  asm volatile("s_wait_asynccnt 0x0" ::: "memory");
}

// Async DMA: 32 lanes x 16B = 512 bytes global -> LDS, tracked by ASYNCcnt.
__device__ __forceinline__ void async_copy_b128(unsigned lds_byte_addr,
                                                const float* gsrc) {
  asm volatile("global_load_async_to_lds_b128 %0, %1, off"
               :: "v"(lds_byte_addr), "v"(gsrc) : "memory");
}

__device__ __forceinline__ int imin(int a, int b) { return a < b ? a : b; }

// -------------------------------------------------------------------------
// GEMM: out[128 x N] = A[128 x K] @ W[K x N] + bias
// mode 0: plain row-major store to `out`
// mode 1: scatter qkv columns into q/k/v workspaces laid out [B,H,T,D]
// Block: 128 threads (4 waves); each wave computes one 16x16 N-tile;
// grid = (N/64, 128/16). A tile staged through LDS, K consumed 64 at a time.
// -------------------------------------------------------------------------
__global__ __launch_bounds__(128)
void gemm128_wmma(const float* __restrict__ A,
                  const float* __restrict__ W,
                  const float* __restrict__ bias,
                  float* __restrict__ out,
                  float* __restrict__ qws,
                  float* __restrict__ kws,
                  float* __restrict__ vws,
                  int K, int N, int mode) {
  __shared__ float As[16 * 68];  // 16 rows x 64 K, padded stride 68

  const int tid   = threadIdx.x;
  const int lane  = tid & 31;
  const int wave  = tid >> 5;
  const int m0    = blockIdx.y * 16;
  const int n0    = blockIdx.x * 64 + wave * 16;
  const int ml    = lane & 15;            // M (for A) / N (for B) within tile
  const int kh    = (lane >> 4) << 1;     // K sub-offset per lane half
  const int thalf = (lane >> 4) << 3;     // C/D row offset per lane half

  v8f acc = {};

  for (int kk0 = 0; kk0 < K; kk0 += 64) {
    __syncthreads();
    {
      // 128 threads load 16x64 floats, 8 contiguous each
      int row = tid >> 3;
      int col = (tid & 7) * 8;
      const float* src = A + (size_t)(m0 + row) * K + kk0 + col;
      #pragma unroll
      for (int j = 0; j < 8; ++j) As[row * 68 + col + j] = src[j];
    }
    __syncthreads();

    // prefetch next K-slab of the weight stream (dominant traffic here)
    if (kk0 + 64 < K)
      __builtin_prefetch(W + (size_t)(kk0 + 64 + kh) * N + n0 + ml, 0, 1);

    #pragma unroll
    for (int k = 0; k < 64; k += 4) {
      v2f a, b;
      a.x = As[ml * 68 + k + kh];
      a.y = As[ml * 68 + k + kh + 1];
      const float* wp = W + (size_t)(kk0 + k + kh) * N + n0 + ml;
      b.x = wp[0];
      b.y = wp[N];
      acc = wmma4(a, b, acc);
    }
  }

  // Epilogue: C/D layout -> element (M = r + thalf, N = ml) in VGPR r
  #pragma unroll
  for (int r = 0; r < 8; ++r) {
    int m = m0 + r + thalf;
    int n = n0 + ml;
    float val = acc[r] + bias[n];
    if (mode == 0) {
      out[(size_t)m * N + n] = val;
    } else {
      int sec = n >> 11;       // 0=q, 1=k, 2=v
      int cc  = n & 2047;
      int h   = cc >> 7;
      int d   = cc & 127;
      int bi  = m >> 3;        // m = b*T + t, T = 8
      int t   = m & 7;
      float* dst = (sec == 0) ? qws : (sec == 1) ? kws : vws;
      dst[(((size_t)bi * H_ + h) * T_ + t) * D_ + d] = val;
    }
  }
}

// -------------------------------------------------------------------------
// Flash attention over the KV cache. One block per (b,h); NW waves each own
// s-chunks ci = wave, wave+NW, ... (16 positions per chunk). K and V chunk
// tiles are DMA'd into LDS with global_load_async_to_lds_b128 (ASYNCcnt),
// WMMA fragments are then read from LDS (conflict-free padded rows).
// New K/V rows (s >= start_pos) come from workspace, cache is read-only.
// -------------------------------------------------------------------------
#define NW 2          // waves per block
#define KVPAD 132     // floats per staged row (128 + 4 -> bank spread)

__global__ __launch_bounds__(32 * NW)
void attn_flash_wmma(const float* __restrict__ qws,
                     const float* __restrict__ kws,
                     const float* __restrict__ vws,
                     const float* __restrict__ k_cache,
                     const float* __restrict__ v_cache,
                     const int* __restrict__ start_p,
                     const int* __restrict__ causal_p,
                     float* __restrict__ y_out) {
  __shared__ __align__(16) float kt[NW][16 * KVPAD];  // staged K tiles
  __shared__ __align__(16) float vt[NW][16 * KVPAD];  // staged V tiles
  __shared__ float st[NW][16 * 17];      // per-wave 16x16 score/P tile
  __shared__ float albuf[NW * 16];       // per-wave per-row rescale factor
  __shared__ float mbuf[NW * 16];        // per-wave row max
  __shared__ float lbuf[NW * 16];        // per-wave row sum
  __shared__ float ybuf[NW * 8 * 128];   // per-wave partial outputs

  const int tid    = threadIdx.x;
  const int lane   = tid & 31;
  const int wave   = tid >> 5;
  const int bh     = blockIdx.x;               // b*H + h
  const int start  = *start_p;
  const int causal = *causal_p;
  const int S      = start + T_;
  const int nchunks = (S + 15) >> 4;

  const int ml    = lane & 15;
  const int kh    = (lane >> 4) << 1;
  const int thalf = (lane >> 4) << 3;
  const float scale = 0.08838834764831845f;    // 1/sqrt(128)

  // Preload Q in A-fragment layout, pre-scaled. Rows t>=8 are zero padding.
  v2f qf[32];
  {
    const float* qrow = qws + ((size_t)bh * T_ + ml) * D_;
    #pragma unroll
    for (int j = 0; j < 32; ++j) {
      float qx = 0.f, qy = 0.f;
      if (ml < T_) { qx = qrow[j * 4 + kh]; qy = qrow[j * 4 + kh + 1]; }
      v2f q; q.x = qx * scale; q.y = qy * scale;
      qf[j] = q;
    }
  }

  v8f acc[8] = {};                       // 16 rows x 128 cols partial output
  float row_m = -INFINITY, row_l = 0.f;  // online softmax state (lanes 0..15)
  float* stw = st[wave];
  float* ktw = kt[wave];
  float* vtw = vt[wave];

  for (int ci = wave; ci < nchunks; ci += NW) {
    const int s0 = ci << 4;

    // Make sure last chunk's ds_load fragment reads drained before the async
    // queue (separately ordered) overwrites the tiles.
    lds_fence();

    // ---- DMA this chunk's K/V tiles into LDS (32 x b128 async copies) ----
    #pragma unroll
    for (int r = 0; r < 16; ++r) {
      int s = s0 + r;
      const float* ksrc = (s < start)
          ? k_cache + ((size_t)bh * MAXS_ + s) * D_
          : kws + ((size_t)bh * T_ + imin(s - start, T_ - 1)) * D_;
      const float* vsrc = (s < start)
          ? v_cache + ((size_t)bh * MAXS_ + s) * D_
          : vws + ((size_t)bh * T_ + imin(s - start, T_ - 1)) * D_;
      unsigned kdst = (unsigned)(size_t)(ktw + r * KVPAD) + (unsigned)(lane * 16);
      unsigned vdst = (unsigned)(size_t)(vtw + r * KVPAD) + (unsigned)(lane * 16);
      async_copy_b128(kdst, ksrc + lane * 4);
      async_copy_b128(vdst, vsrc + lane * 4);
    }
    async_fence();

    // ---- scores tile = Q @ K^T (K-dim = D = 128, 32 WMMAs) ----
    v8f sc = {};
    #pragma unroll
    for (int j = 0; j < 32; ++j) {
      v2f b;
      b.x = ktw[ml * KVPAD + j * 4 + kh];
      b.y = ktw[ml * KVPAD + j * 4 + kh + 1];
      sc = wmma4(qf[j], b, sc);
    }

    // ---- stage scores to LDS (C-layout -> row-major) ----
    #pragma unroll
    for (int r = 0; r < 8; ++r) stw[(r + thalf) * 17 + ml] = sc[r];
    lds_fence();

    // ---- online softmax, one row per lane (lanes 0..15) ----
    if (lane < 16) {
      const int t = lane;
      float cmax = -INFINITY;
      float pv[16];
      #pragma unroll
      for (int j = 0; j < 16; ++j) {
        int s = s0 + j;
        bool ok = (s < S) && (!causal || s <= t);
        float v = ok ? stw[t * 17 + j] : -INFINITY;
        pv[j] = v;
        cmax = fmaxf(cmax, v);
      }
      float nm = fmaxf(row_m, cmax);
      float alpha = 1.f, lsum = 0.f;
      if (nm != -INFINITY) {
        alpha = __expf(row_m - nm);  // row_m = -inf -> alpha = 0
        #pragma unroll
        for (int j = 0; j < 16; ++j) {
          float p = (pv[j] == -INFINITY) ? 0.f : __expf(pv[j] - nm);
          stw[t * 17 + j] = p;
          lsum += p;
        }
      } else {
        #pragma unroll
        for (int j = 0; j < 16; ++j) stw[t * 17 + j] = 0.f;
      }
      row_l = row_l * alpha + lsum;
      row_m = nm;
      albuf[wave * 16 + t] = alpha;
    }
    lds_fence();

    // ---- rescale running accumulators by per-row alpha ----
    #pragma unroll
    for (int r = 0; r < 8; ++r) {
      float av = albuf[wave * 16 + r + thalf];
      #pragma unroll
      for (int j = 0; j < 8; ++j) acc[j][r] = acc[j][r] * av;
    }

    // ---- acc += P @ V (K-dim = 16 positions, 4 WMMAs x 8 d-tiles) ----
    v2f pf[4];
    #pragma unroll
    for (int ks = 0; ks < 4; ++ks) {
      pf[ks].x = stw[ml * 17 + ks * 4 + kh];
      pf[ks].y = stw[ml * 17 + ks * 4 + kh + 1];
    }
    #pragma unroll
    for (int ks = 0; ks < 4; ++ks) {
      int srl = ks * 4 + kh;
      #pragma unroll
      for (int j = 0; j < 8; ++j) {
        v2f b;
        b.x = vtw[srl * KVPAD + j * 16 + ml];
        b.y = vtw[(srl + 1) * KVPAD + j * 16 + ml];
        acc[j] = wmma4(pf[ks], b, acc[j]);
      }
    }
  }

  // ---- merge the NW waves' partials (log-sum-exp) ----
  #pragma unroll
  for (int r = 0; r < 8; ++r) {
    int t = r + thalf;
    if (t < 8) {  // lanes 0..15 carry the valid rows
      #pragma unroll
      for (int j = 0; j < 8; ++j)
        ybuf[(wave * 8 + t) * 128 + j * 16 + ml] = acc[j][r];
    }
  }
  if (lane < 16) {
    mbuf[wave * 16 + lane] = row_m;
    lbuf[wave * 16 + lane] = row_l;
  }
  __syncthreads();

  {
    int t = tid >> 3;      // 0..7 (valid rows), 8 threads per row
    int dsub = tid & 7;
    float m0v = mbuf[t], m1v = mbuf[16 + t];
    float M = fmaxf(m0v, m1v);
    float e0 = (m0v == -INFINITY) ? 0.f : __expf(m0v - M);
    float e1 = (m1v == -INFINITY) ? 0.f : __expf(m1v - M);
    float L = lbuf[t] * e0 + lbuf[16 + t] * e1;
    float invL = (L > 0.f) ? 1.f / L : 0.f;
    int b = bh >> 4, h = bh & 15;
    #pragma unroll
    for (int j = 0; j < 16; ++j) {
      int d = j * 8 + dsub;
      float sum = ybuf[t * 128 + d] * e0 + ybuf[(8 + t) * 128 + d] * e1;
      // y laid out [B,T,C] row-major so proj GEMM reads it directly
      y_out[((size_t)b * T_ + t) * C_ + h * D_ + d] = sum * invL;
    }
  }
}

// -------------------------------------------------------------------------
extern "C" void kernel_launch(void* const* d_in, const int* in_sizes, int n_in,
                              void* d_out, int out_size, void* d_ws, size_t ws_size,
                              hipStream_t stream) {
  const float* x       = (const float*)d_in[0];
  const float* k_cache = (const float*)d_in[1];
  const float* v_cache = (const float*)d_in[2];
  const float* w_attn  = (const float*)d_in[3];
  const float* b_attn  = (const float*)d_in[4];
  const float* w_proj  = (const float*)d_in[5];
  const float* b_proj  = (const float*)d_in[6];
  const int*   start_p = (const int*)d_in[7];
  const int*   caus_p  = (const int*)d_in[8];
  float*       out     = (float*)d_out;

  // Workspace: q/k/v new tiles [B,H,T,D] + y [B,T,C]  (4 MB total)
  const size_t qkv_elems = (size_t)B_ * H_ * T_ * D_;  // 262144
  float* qws = (float*)d_ws;
  float* kws = qws + qkv_elems;
  float* vws = kws + qkv_elems;
  float* yws = vws + qkv_elems;

  // 1) QKV projection: [128 x 2048] @ [2048 x 6144] + bias, scatter to q/k/v
  gemm128_wmma<<<dim3(6144 / 64, 128 / 16), 128, 0, stream>>>(
      x, w_attn, b_attn, nullptr, qws, kws, vws, /*K=*/2048, /*N=*/6144, /*mode=*/1);

  // 2) Flash attention over KV cache (+ new rows from workspace)
  attn_flash_wmma<<<dim3(B_ * H_), 32 * NW, 0, stream>>>(
      qws, kws, vws, k_cache, v_cache, start_p, caus_p, yws);

  // 3) Output projection: [128 x 2048] @ [2048 x 2048] + bias -> d_out
  gemm128_wmma<<<dim3(2048 / 64, 128 / 16), 128, 0, stream>>>(
      yws, w_proj, b_proj, out, nullptr, nullptr, nullptr, /*K=*/2048, /*N=*/2048, /*mode=*/0);
}